// TGAT_60979945669280
// MI455X (gfx1250) — compile-verified
//
#include <hip/hip_runtime.h>
#include <hip/hip_bf16.h>

typedef __attribute__((ext_vector_type(2))) float v2f;
typedef __attribute__((ext_vector_type(8))) float v8f;

#define HID 64
#define TDIM 16
#define NEG_SLOPE 0.01f

// ---------------------------------------------------------------------------
// Ordered float <-> uint key (monotonic) for atomicMax-based segment max.
// ---------------------------------------------------------------------------
__device__ __forceinline__ unsigned f2key(float f) {
    unsigned u = __float_as_uint(f);
    return (u & 0x80000000u) ? ~u : (u | 0x80000000u);
}
__device__ __forceinline__ float key2f(unsigned k) {
    return __uint_as_float((k & 0x80000000u) ? (k & 0x7FFFFFFFu) : ~k);
}

// ---------------------------------------------------------------------------
// H[N,64] = X[N,K] @ W[K,64] + bias, plus fused attention projections:
//   s_dst[n] = H[n,:] . attn_w[0:64],  s_src[n] = H[n,:] . attn_w[64:128]
// One wave per 16-row tile; 4 x v_wmma_f32_16x16x4_f32 per K-step covers
// all 64 output columns in FP32 (full precision, zero accuracy risk).
// ---------------------------------------------------------------------------
__global__ __launch_bounds__(256) void gemm_wmma_f32(
    const float* __restrict__ X, const float* __restrict__ W,
    const float* __restrict__ bias, const float* __restrict__ attn_w,
    float* __restrict__ H, float* __restrict__ s_dst, float* __restrict__ s_src,
    int nrows, int K)
{
    int wave = (int)((blockIdx.x * blockDim.x + threadIdx.x) >> 5);
    int lane = (int)(threadIdx.x & 31);
    int row0 = wave * 16;
    if (row0 >= nrows) return;

    int m    = lane & 15;        // A: row within tile
    int n    = lane & 15;        // B/C/D: column within 16-wide tile
    int kof  = (lane >> 4) * 2;  // A/B: K offset for lane half
    int half = lane >> 4;        // C/D: M half (rows 0-7 vs 8-15)

    v8f acc[4] = {v8f{}, v8f{}, v8f{}, v8f{}};

    const float* xrow = X + (size_t)(row0 + m) * K;
    for (int k0 = 0; k0 < K; k0 += 4) {
        v2f a;
        a.x = xrow[k0 + kof];
        a.y = xrow[k0 + kof + 1];
#pragma unroll
        for (int t = 0; t < 4; ++t) {
            v2f b;
            b.x = W[(size_t)(k0 + kof)     * HID + t * 16 + n];
            b.y = W[(size_t)(k0 + kof + 1) * HID + t * 16 + n];
            acc[t] = __builtin_amdgcn_wmma_f32_16x16x4_f32(
                false, a, false, b, (short)0, acc[t], false, false);
        }
    }

    // epilogue: + bias, store H, fused attention scalar projections
    float wa[4], wb[4];
#pragma unroll
    for (int t = 0; t < 4; ++t) {
        float bv = bias[t * 16 + n];
        wa[t] = attn_w[t * 16 + n];        // dst-half of attention weight
        wb[t] = attn_w[HID + t * 16 + n];  // src-half
#pragma unroll
        for (int i = 0; i < 8; ++i) acc[t][i] += bv;
    }

#pragma unroll
    for (int i = 0; i < 8; ++i) {
        size_t row = (size_t)(row0 + half * 8 + i);
#pragma unroll
        for (int t = 0; t < 4; ++t)
            H[row * HID + t * 16 + n] = acc[t][i];
    }

#pragma unroll
    for (int i = 0; i < 8; ++i) {
        float pd = 0.f, ps = 0.f;
#pragma unroll
        for (int t = 0; t < 4; ++t) { pd += acc[t][i] * wa[t]; ps += acc[t][i] * wb[t]; }
        // reduce over the 16 lanes that share this row (xor stays within half)
#pragma unroll
        for (int off = 8; off; off >>= 1) {
            pd += __shfl_xor(pd, off, 32);
            ps += __shfl_xor(ps, off, 32);
        }
        if (n == 0) {
            int row = row0 + half * 8 + i;
            s_dst[row] = pd;
            s_src[row] = ps;
        }
    }
}

// ---------------------------------------------------------------------------
// Pass 1: per-edge raw attention logit (leaky-relu) + segment max (atomic).
// ---------------------------------------------------------------------------
__global__ __launch_bounds__(256) void edge_alpha_kernel(
    const int* __restrict__ src, const int* __restrict__ dst,
    const float* __restrict__ etime,
    const float* __restrict__ s_dst, const float* __restrict__ s_src,
    const float* __restrict__ attn_w, const float* __restrict__ attn_b,
    const float* __restrict__ t_w, const float* __restrict__ t_b,
    float* __restrict__ alpha, unsigned* __restrict__ nmax, int E)
{
    int e = (int)(blockIdx.x * blockDim.x + threadIdx.x);
    if (e >= E) return;
    float t = etime[e];
    float c = 0.f;
#pragma unroll
    for (int k = 0; k < TDIM; ++k)
        c += sinf(t * t_w[k] + t_b[k]) * attn_w[2 * HID + k];
    int d = dst[e];
    float a = s_dst[d] + s_src[src[e]] + c + attn_b[0];
    a = (a > 0.f) ? a : NEG_SLOPE * a;
    alpha[e] = a;
    atomicMax(&nmax[d], f2key(a));
}

// ---------------------------------------------------------------------------
// Pass 2: exp(a - segmax) + segment sum (atomic).
// ---------------------------------------------------------------------------
__global__ __launch_bounds__(256) void edge_exp_kernel(
    const int* __restrict__ dst, float* __restrict__ alpha,
    const unsigned* __restrict__ nmax, float* __restrict__ nsum, int E)
{
    int e = (int)(blockIdx.x * blockDim.x + threadIdx.x);
    if (e >= E) return;
    int d = dst[e];
    float ex = expf(alpha[e] - key2f(nmax[d]));
    alpha[e] = ex;
    atomicAdd(&nsum[d], ex);
}

// ---------------------------------------------------------------------------
// Pass 3: normalized message scatter. 64 threads per edge -> fully coalesced
// 256B gather of h[src] and 256B atomic-add burst into AGG[dst].
// ---------------------------------------------------------------------------
__global__ __launch_bounds__(256) void edge_scatter_kernel(
    const int* __restrict__ src, const int* __restrict__ dst,
    const float* __restrict__ alpha, const float* __restrict__ nsum,
    const float* __restrict__ H, float* __restrict__ AGG, int E)
{
    unsigned gid = blockIdx.x * blockDim.x + threadIdx.x;
    int e = (int)(gid >> 6);
    int c = (int)(gid & 63u);
    if (e >= E) return;
    int d = dst[e];
    int s = src[e];
    float an = alpha[e] / (nsum[d] + 1e-16f);
    atomicAdd(&AGG[(size_t)d * HID + c], H[(size_t)s * HID + c] * an);
}

__global__ __launch_bounds__(256) void relu_kernel(float* __restrict__ v, size_t n)
{
    size_t i = (size_t)blockIdx.x * blockDim.x + threadIdx.x;
    if (i < n) v[i] = fmaxf(v[i], 0.f);
}

// ---------------------------------------------------------------------------
// Final classifier: out[n,0:2] = AGG[n,:] @ cls_w[64,2] + cls_b
// ---------------------------------------------------------------------------
__global__ __launch_bounds__(256) void classify_kernel(
    const float* __restrict__ AGG, const float* __restrict__ cls_w,
    const float* __restrict__ cls_b, float* __restrict__ out, int N)
{
    int nd = (int)(blockIdx.x * blockDim.x + threadIdx.x);
    if (nd >= N) return;
    float a0 = cls_b[0], a1 = cls_b[1];
    const float* row = AGG + (size_t)nd * HID;
#pragma unroll 4
    for (int c = 0; c < HID; ++c) {
        float h = row[c];
        a0 += h * cls_w[2 * c];
        a1 += h * cls_w[2 * c + 1];
    }
    out[2 * (size_t)nd]     = a0;
    out[2 * (size_t)nd + 1] = a1;
}

// ---------------------------------------------------------------------------
extern "C" void kernel_launch(void* const* d_in, const int* in_sizes, int n_in,
                              void* d_out, int out_size, void* d_ws, size_t ws_size,
                              hipStream_t stream)
{
    const float* x       = (const float*)d_in[0];
    const int*   eidx    = (const int*)  d_in[1];
    const float* etime   = (const float*)d_in[2];
    const float* lin1_w  = (const float*)d_in[3];
    const float* lin1_b  = (const float*)d_in[4];
    const float* attn1_w = (const float*)d_in[5];
    const float* attn1_b = (const float*)d_in[6];
    const float* t1_w    = (const float*)d_in[7];
    const float* t1_b    = (const float*)d_in[8];
    const float* lin2_w  = (const float*)d_in[9];
    const float* lin2_b  = (const float*)d_in[10];
    const float* attn2_w = (const float*)d_in[11];
    const float* attn2_b = (const float*)d_in[12];
    const float* t2_w    = (const float*)d_in[13];
    const float* t2_b    = (const float*)d_in[14];
    const float* cls_w   = (const float*)d_in[15];
    const float* cls_b   = (const float*)d_in[16];

    const int N = in_sizes[0] / 128;   // 100000 (multiple of 16)
    const int E = in_sizes[2];         // 1600000
    const int IN_DIM = in_sizes[0] / N;

    const int* srcv = eidx;        // edge_index[0]
    const int* dstv = eidx + E;    // edge_index[1]

    // workspace carve-up
    float*    H     = (float*)d_ws;                    // N*64
    float*    AGG   = H + (size_t)N * HID;             // N*64
    float*    sD    = AGG + (size_t)N * HID;           // N
    float*    sS    = sD + N;                          // N
    float*    alpha = sS + N;                          // E
    unsigned* nmax  = (unsigned*)(alpha + E);          // N
    float*    nsum  = (float*)(nmax + N);              // N

    const int tiles      = N / 16;
    const int gemmBlocks = (tiles + 7) / 8;            // 8 waves per 256-thread block
    const int eBlocks    = (E + 255) / 256;
    const int sBlocks    = (int)(((size_t)E * HID + 255) / 256);
    const int vBlocks    = (int)(((size_t)N * HID + 255) / 256);
    const int nBlocks    = (N + 255) / 256;

    // ---------------- Layer 1 ----------------
    gemm_wmma_f32<<<gemmBlocks, 256, 0, stream>>>(
        x, lin1_w, lin1_b, attn1_w, H, sD, sS, N, IN_DIM);

    hipMemsetAsync(nmax, 0, (size_t)N * sizeof(unsigned), stream);
    hipMemsetAsync(nsum, 0, (size_t)N * sizeof(float), stream);
    hipMemsetAsync(AGG,  0, (size_t)N * HID * sizeof(float), stream);

    edge_alpha_kernel<<<eBlocks, 256, 0, stream>>>(
        srcv, dstv, etime, sD, sS, attn1_w, attn1_b, t1_w, t1_b, alpha, nmax, E);
    edge_exp_kernel<<<eBlocks, 256, 0, stream>>>(dstv, alpha, nmax, nsum, E);
    edge_scatter_kernel<<<sBlocks, 256, 0, stream>>>(
        srcv, dstv, alpha, nsum, H, AGG, E);

    relu_kernel<<<vBlocks, 256, 0, stream>>>(AGG, (size_t)N * HID);

    // ---------------- Layer 2 ----------------
    gemm_wmma_f32<<<gemmBlocks, 256, 0, stream>>>(
        AGG, lin2_w, lin2_b, attn2_w, H, sD, sS, N, HID);

    hipMemsetAsync(nmax, 0, (size_t)N * sizeof(unsigned), stream);
    hipMemsetAsync(nsum, 0, (size_t)N * sizeof(float), stream);
    hipMemsetAsync(AGG,  0, (size_t)N * HID * sizeof(float), stream);

    edge_alpha_kernel<<<eBlocks, 256, 0, stream>>>(
        srcv, dstv, etime, sD, sS, attn2_w, attn2_b, t2_w, t2_b, alpha, nmax, E);
    edge_exp_kernel<<<eBlocks, 256, 0, stream>>>(dstv, alpha, nmax, nsum, E);
    edge_scatter_kernel<<<sBlocks, 256, 0, stream>>>(
        srcv, dstv, alpha, nsum, H, AGG, E);

    // ---------------- Classifier ----------------
    classify_kernel<<<nBlocks, 256, 0, stream>>>(AGG, cls_w, cls_b, (float*)d_out, N);
}